// WeightVolumeIndex_22376779612534
// MI455X (gfx1250) — compile-verified
//
#include <hip/hip_runtime.h>
#include <stdint.h>

#define NGRID 128
#define BLOCK 256

typedef __attribute__((ext_vector_type(4))) int   v4i;
typedef __attribute__((ext_vector_type(4))) float v4f;

// ---------------------------------------------------------------------------
// CDNA5 async global->LDS copy (16 bytes per active lane), tracked by ASYNCcnt.
// vdst = LDS byte address (low 32 bits of the generic shared pointer),
// vaddr = 64-bit global byte address, GV mode (no SADDR -> "off").
// ---------------------------------------------------------------------------
__device__ __forceinline__ void async_g2lds_b128(unsigned lds_addr,
                                                 unsigned long long gaddr) {
    asm volatile("global_load_async_to_lds_b128 %0, %1, off"
                 :
                 : "v"(lds_addr), "v"(gaddr)
                 : "memory");
}

__device__ __forceinline__ void wait_asynccnt0() {
    asm volatile("s_wait_asynccnt 0" ::: "memory");
}

__global__ __launch_bounds__(BLOCK) void WeightVolumeIndex_22376779612534_kernel(
    const float* __restrict__ points,    // [M,3]
    const float* __restrict__ bbox_min,  // [3]
    const float* __restrict__ bbox_max,  // [3]
    const int*   __restrict__ grid,      // [128,128,128] (z,y,x)
    int M,
    int*   __restrict__ out_ids,         // [M,8]
    float* __restrict__ out_w)           // [M,8]
{
    __shared__ float sp[BLOCK * 3];

    const int       tid  = threadIdx.x;
    const long long base = (long long)blockIdx.x * BLOCK;
    const long long npts = (M - base < BLOCK) ? (M - base) : BLOCK;

    // ---- Stage this block's points (npts*12 bytes) into LDS asynchronously.
    // 256 points * 12 B = 3072 B = 192 lanes * 16 B (waves 0..5 active).
    const long long nbytes = npts * 12;
    if ((long long)tid * 16 < nbytes) {
        unsigned lds_addr =
            (unsigned)(uintptr_t)(&sp[0]) + (unsigned)(tid * 16);
        unsigned long long gaddr =
            (unsigned long long)(uintptr_t)(points + base * 3) +
            (unsigned long long)(tid * 16);
        async_g2lds_b128(lds_addr, gaddr);
    }
    wait_asynccnt0();   // each wave waits for its own async transfers
    __syncthreads();    // then publish LDS to the whole workgroup

    if (tid < npts) {
        const float px = sp[tid * 3 + 0];
        const float py = sp[tid * 3 + 1];
        const float pz = sp[tid * 3 + 2];

        const float bx0 = bbox_min[0], by0 = bbox_min[1], bz0 = bbox_min[2];
        const float bx1 = bbox_max[0], by1 = bbox_max[1], bz1 = bbox_max[2];

        // normalize to [-1,1] exactly like the reference
        const float ux = (px - bx0) / (bx1 - bx0) * 2.0f - 1.0f;
        const float uy = (py - by0) / (by1 - by0) * 2.0f - 1.0f;
        const float uz = (pz - bz0) / (bz1 - bz0) * 2.0f - 1.0f;

        const float scale = 0.5f * (float)(NGRID - 1);
        const float cx = (ux + 1.0f) * scale;
        const float cy = (uy + 1.0f) * scale;
        const float cz = (uz + 1.0f) * scale;

        int ix0 = (int)floorf(cx);
        int iy0 = (int)floorf(cy);
        int iz0 = (int)floorf(cz);
        ix0 = ix0 < 0 ? 0 : (ix0 > NGRID - 2 ? NGRID - 2 : ix0);
        iy0 = iy0 < 0 ? 0 : (iy0 > NGRID - 2 ? NGRID - 2 : iy0);
        iz0 = iz0 < 0 ? 0 : (iz0 > NGRID - 2 ? NGRID - 2 : iz0);

        float fx = cx - (float)ix0;
        float fy = cy - (float)iy0;
        float fz = cz - (float)iz0;
        fx = fminf(fmaxf(fx, 0.0f), 1.0f);
        fy = fminf(fmaxf(fy, 0.0f), 1.0f);
        fz = fminf(fmaxf(fz, 0.0f), 1.0f);

        // 8 corner gathers; grid is L2-resident (8 MB << 192 MB L2)
        const int* g00 = grid + (((int)iz0 * NGRID + iy0) * NGRID + ix0);
        const int* g01 = g00 + NGRID;
        const int* g10 = g00 + NGRID * NGRID;
        const int* g11 = g10 + NGRID;
        const int i000 = g00[0], i001 = g00[1];
        const int i010 = g01[0], i011 = g01[1];
        const int i100 = g10[0], i101 = g10[1];
        const int i110 = g11[0], i111 = g11[1];

        const float fx0 = 1.0f - fx, fy0 = 1.0f - fy, fz0 = 1.0f - fz;
        const float w00 = fz0 * fy0;   // dz=0, dy=0
        const float w01 = fz0 * fy;    // dz=0, dy=1
        const float w10 = fz  * fy0;   // dz=1, dy=0
        const float w11 = fz  * fy;    // dz=1, dy=1

        const long long o = (base + tid) * 8;

        // ids: two 128-bit non-temporal stores (don't pollute L2)
        v4i ia = {i000, i001, i010, i011};
        v4i ib = {i100, i101, i110, i111};
        __builtin_nontemporal_store(ia, (v4i*)(out_ids + o));
        __builtin_nontemporal_store(ib, (v4i*)(out_ids + o) + 1);

        // weights in reference (dz,dy,dx) order
        v4f wa = {w00 * fx0, w00 * fx, w01 * fx0, w01 * fx};
        v4f wb = {w10 * fx0, w10 * fx, w11 * fx0, w11 * fx};
        __builtin_nontemporal_store(wa, (v4f*)(out_w + o));
        __builtin_nontemporal_store(wb, (v4f*)(out_w + o) + 1);
    }
}

extern "C" void kernel_launch(void* const* d_in, const int* in_sizes, int n_in,
                              void* d_out, int out_size, void* d_ws, size_t ws_size,
                              hipStream_t stream) {
    const float* points   = (const float*)d_in[0];
    const float* bbox_min = (const float*)d_in[1];
    const float* bbox_max = (const float*)d_in[2];
    const int*   grid_id  = (const int*)d_in[3];

    const int M = in_sizes[0] / 3;

    int*   out_ids = (int*)d_out;
    float* out_w   = (float*)d_out + (size_t)M * 8;

    const int blocks = (M + BLOCK - 1) / BLOCK;
    WeightVolumeIndex_22376779612534_kernel<<<blocks, BLOCK, 0, stream>>>(
        points, bbox_min, bbox_max, grid_id, M, out_ids, out_w);
}